// AuxNetwork_18571438588238
// MI455X (gfx1250) — compile-verified
//
#include <hip/hip_runtime.h>

typedef float v2f __attribute__((ext_vector_type(2)));
typedef float v8f __attribute__((ext_vector_type(8)));

#define LN_EPS 1e-5f
#define LRELU_SLOPE 0.01f

// ---------------------------------------------------------------------------
// CDNA5 async global->LDS copy (16B per lane), tracked by ASYNCcnt.
// ldsOff = wave-relative LDS byte offset (low 32 bits of generic shared ptr).
// ---------------------------------------------------------------------------
__device__ __forceinline__ void async_copy_b128(unsigned ldsOff, const float* g)
{
    asm volatile("global_load_async_to_lds_b128 %0, %1, off"
                 :: "v"(ldsOff), "v"(g)
                 : "memory");
}

// ---------------------------------------------------------------------------
// Generic tiled GEMM:  C[M,N] = A[M,K] @ W[N,K]^T + bias[N]
//   Template WM = per-wave M-tile (16 or 32). Block tile = (4*WM) x 64.
//   grid.x = N/64, grid.y = M/(4*WM), grid.z = batch
//   Requires: K % 32 == 0, N % 64 == 0, M % (4*WM) == 0.
// 8 wave32 waves: wave (wid&3) -> M sub-tile, (wid>>2) -> 32-wide N sub-tile.
// WM=32: 4 accumulators (32x32 per wave) -> 4 V_WMMA_F32_16X16X4_F32 per
// 4 LDS fragment reads; double-buffered LDS filled by async-to-LDS copies.
// ---------------------------------------------------------------------------
template <int WM>
__global__ __launch_bounds__(256) void gemm_wmma_f32(
    const float* __restrict__ A, long lda, long strideA,
    const float* __restrict__ W, long ldw, long strideW,
    const float* __restrict__ bias, long strideBias,
    float* __restrict__ C, long ldc, long strideC,
    int K)
{
    constexpr int BM = WM * 4;            // 64 or 128
    constexpr int SL = BM / 32 + 2;       // async issues per thread per slab

    // 36-float row pitch: conflict-free fragment reads + 16B-aligned rows
    __shared__ float As[2][BM][36];
    __shared__ float Ws[2][64][36];

    const int tid  = threadIdx.x;
    const int lane = tid & 31;
    const int wid  = tid >> 5;
    const int l    = lane & 15;   // M index (A frag) / N index (B frag)
    const int hi   = lane >> 4;   // selects K pair {k,k+1} vs {k+2,k+3}
    const int m0   = (wid & 3) * WM;
    const int n0   = (wid >> 2) * 32;

    const long mbase = (long)blockIdx.y * BM;
    const long nbase = (long)blockIdx.x * 64;

    const float* Ab = A    + (long)blockIdx.z * strideA;
    const float* Wb = W    + (long)blockIdx.z * strideW;
    const float* Bb = bias + (long)blockIdx.z * strideBias;
    float*       Cb = C    + (long)blockIdx.z * strideC;

    // Per-thread fill assignment: flat float4 id f -> (row, 4-col)
    auto issue = [&](int slab) {
        const int  buf = slab & 1;
        const long k0  = (long)slab * 32;
#pragma unroll
        for (int j = 0; j < BM / 32; ++j) {
            const int f   = tid + j * 256;
            const int row = f >> 3;
            const int c4  = (f & 7) << 2;
            async_copy_b128((unsigned)(size_t)&As[buf][row][c4],
                            Ab + (mbase + row) * lda + k0 + c4);
        }
#pragma unroll
        for (int j = 0; j < 2; ++j) {
            const int f   = tid + j * 256;
            const int row = f >> 3;
            const int c4  = (f & 7) << 2;
            async_copy_b128((unsigned)(size_t)&Ws[buf][row][c4],
                            Wb + (nbase + row) * ldw + k0 + c4);
        }
    };

    v8f acc00 = {}, acc01 = {}, acc10 = {}, acc11 = {};

    const int nslabs = K >> 5;
    issue(0);
    for (int s = 0; s < nslabs; ++s) {
        const int buf = s & 1;
        if (s + 1 < nslabs) {
            issue(s + 1);
            // Loads complete in order: wait until only slab s+1's SL remain.
            if constexpr (SL == 6)
                asm volatile("s_wait_asynccnt 0x6" ::: "memory");
            else
                asm volatile("s_wait_asynccnt 0x4" ::: "memory");
        } else {
            asm volatile("s_wait_asynccnt 0x0" ::: "memory");
        }
        __syncthreads();

#pragma unroll
        for (int kk = 0; kk < 32; kk += 4) {
            // A frags (16x4): lanes 0-15 K={kk,kk+1}, lanes 16-31 K={kk+2,kk+3}
            const v2f a0 = *(const v2f*)(&As[buf][m0 + l][kk + 2 * hi]);
            // B frags (4x16): lane -> N column, VGPR pair -> same K pairing
            const v2f b0 = *(const v2f*)(&Ws[buf][n0 + l][kk + 2 * hi]);
            const v2f b1 = *(const v2f*)(&Ws[buf][n0 + 16 + l][kk + 2 * hi]);
            acc00 = __builtin_amdgcn_wmma_f32_16x16x4_f32(
                false, a0, false, b0, (short)0, acc00, false, false);
            acc01 = __builtin_amdgcn_wmma_f32_16x16x4_f32(
                false, a0, false, b1, (short)0, acc01, false, false);
            if constexpr (WM == 32) {
                const v2f a1 = *(const v2f*)(&As[buf][m0 + 16 + l][kk + 2 * hi]);
                acc10 = __builtin_amdgcn_wmma_f32_16x16x4_f32(
                    false, a1, false, b0, (short)0, acc10, false, false);
                acc11 = __builtin_amdgcn_wmma_f32_16x16x4_f32(
                    false, a1, false, b1, (short)0, acc11, false, false);
            }
        }
        __syncthreads();
    }

    // C/D layout: VGPR r -> M = tileRow + r + 8*hi, N = tileCol + l
    auto store_tile = [&](const v8f& acc, long mrow, long ncol) {
#pragma unroll
        for (int r = 0; r < 8; ++r) {
            const long row = mrow + r + 8 * hi;
            const long col = ncol + l;
            Cb[row * ldc + col] = acc[r] + Bb[col];
        }
    };
    store_tile(acc00, mbase + m0, nbase + n0);
    store_tile(acc01, mbase + m0, nbase + n0 + 16);
    if constexpr (WM == 32) {
        store_tile(acc10, mbase + m0 + 16, nbase + n0);
        store_tile(acc11, mbase + m0 + 16, nbase + n0 + 16);
    }
}

// ---------------------------------------------------------------------------
// Row LayerNorm + LeakyReLU. One 256-thread block per row. in==out is safe.
// ---------------------------------------------------------------------------
__global__ __launch_bounds__(256) void ln_lrelu_kernel(
    const float* __restrict__ in, float* __restrict__ out,
    const float* __restrict__ g, const float* __restrict__ b, int N)
{
    __shared__ float s0[256];
    __shared__ float s1[256];
    const long row = blockIdx.x;
    const float* x = in + row * (long)N;
    float sum = 0.f, sq = 0.f;
    for (int c = threadIdx.x; c < N; c += 256) {
        const float v = x[c];
        sum += v;
        sq  += v * v;
    }
    s0[threadIdx.x] = sum;
    s1[threadIdx.x] = sq;
    __syncthreads();
    for (int off = 128; off > 0; off >>= 1) {
        if ((int)threadIdx.x < off) {
            s0[threadIdx.x] += s0[threadIdx.x + off];
            s1[threadIdx.x] += s1[threadIdx.x + off];
        }
        __syncthreads();
    }
    const float mean = s0[0] / (float)N;
    const float var  = s1[0] / (float)N - mean * mean;
    const float inv  = rsqrtf(var + LN_EPS);
    float* y = out + row * (long)N;
    for (int c = threadIdx.x; c < N; c += 256) {
        const float v = (x[c] - mean) * inv * g[c] + b[c];
        y[c] = (v >= 0.f) ? v : LRELU_SLOPE * v;
    }
}

// ---------------------------------------------------------------------------
// Final tiny projection: out[m, 0..2] = in[m, :] @ w[0..2, :]^T + b. K = 1024.
// ---------------------------------------------------------------------------
__global__ __launch_bounds__(256) void fc4_kernel(
    const float* __restrict__ in, const float* __restrict__ w,
    const float* __restrict__ b, float* __restrict__ out, int K)
{
    __shared__ float red[3][256];
    const long row = blockIdx.x;
    const float* x = in + row * (long)K;
    float a0 = 0.f, a1 = 0.f, a2 = 0.f;
    for (int k = threadIdx.x; k < K; k += 256) {
        const float v = x[k];
        a0 += v * w[k];
        a1 += v * w[K + k];
        a2 += v * w[2 * K + k];
    }
    red[0][threadIdx.x] = a0;
    red[1][threadIdx.x] = a1;
    red[2][threadIdx.x] = a2;
    __syncthreads();
    for (int off = 128; off > 0; off >>= 1) {
        if ((int)threadIdx.x < off) {
            red[0][threadIdx.x] += red[0][threadIdx.x + off];
            red[1][threadIdx.x] += red[1][threadIdx.x + off];
            red[2][threadIdx.x] += red[2][threadIdx.x + off];
        }
        __syncthreads();
    }
    if (threadIdx.x < 3)
        out[row * 3 + threadIdx.x] = red[threadIdx.x][0] + b[threadIdx.x];
}

// ---------------------------------------------------------------------------
extern "C" void kernel_launch(void* const* d_in, const int* in_sizes, int n_in,
                              void* d_out, int out_size, void* d_ws, size_t ws_size,
                              hipStream_t stream)
{
    (void)in_sizes; (void)n_in; (void)out_size; (void)ws_size;

    const float* x        = (const float*)d_in[0];
    const float* w1       = (const float*)d_in[1];
    const float* b1       = (const float*)d_in[2];
    const float* ln1_g    = (const float*)d_in[3];
    const float* ln1_b    = (const float*)d_in[4];
    const float* fc2_w    = (const float*)d_in[5];
    const float* fc2_b    = (const float*)d_in[6];
    const float* ln2_g    = (const float*)d_in[7];
    const float* ln2_b    = (const float*)d_in[8];
    const float* blk_w    = (const float*)d_in[9];
    const float* blk_b    = (const float*)d_in[10];
    const float* blk_ln_g = (const float*)d_in[11];
    const float* blk_ln_b = (const float*)d_in[12];
    const float* fc3_w    = (const float*)d_in[13];
    const float* fc3_b    = (const float*)d_in[14];
    const float* ln3_g    = (const float*)d_in[15];
    const float* ln3_b    = (const float*)d_in[16];
    const float* fc4_w    = (const float*)d_in[17];
    const float* fc4_b    = (const float*)d_in[18];

    constexpr int  Bz = 64, NW = 256, WIN = 512, H = 1024, H1 = 2048, H2 = 1024, L = 2;
    constexpr long M  = (long)Bz * NW;  // 16384 rows

    float* out = (float*)d_out;
    float* o1 = out;                 // [M, H]
    float* o2 = o1 + M * H;          // [M, H1]
    float* o3 = o2 + M * H1;         // [M, H2]
    float* o4 = o3 + M * H2;         // [M, 3]
    float* pre = (float*)d_ws;       // scratch: M*H1 floats (128 MB)

    const dim3 blk(256);

    // Stage 1: per-window batched GEMM [64 x 512] @ [512 x 1024] (256 batches)
    // HBM-bound on w1 (512 MB); BM=64 since M = batch rows = 64.
    gemm_wmma_f32<16><<<dim3(H / 64, Bz / 64, NW), blk, 0, stream>>>(
        x, (long)NW * WIN, (long)WIN,
        w1, (long)WIN, (long)H * WIN,
        b1, (long)H,
        o1, (long)NW * H, (long)H, WIN);
    ln_lrelu_kernel<<<dim3((unsigned)M), blk, 0, stream>>>(o1, o1, ln1_g, ln1_b, H);

    // fc2: [M x H] @ [H x H1]  (weights L2-resident; compute-bound)
    gemm_wmma_f32<32><<<dim3(H1 / 64, M / 128, 1), blk, 0, stream>>>(
        o1, (long)H, 0, fc2_w, (long)H, 0, fc2_b, 0, pre, (long)H1, 0, H);
    ln_lrelu_kernel<<<dim3((unsigned)M), blk, 0, stream>>>(pre, o2, ln2_g, ln2_b, H1);

    // L residual-free blocks: [M x H1] @ [H1 x H1], ping-pong o2 <-> ws
    for (int i = 0; i < L; ++i) {
        gemm_wmma_f32<32><<<dim3(H1 / 64, M / 128, 1), blk, 0, stream>>>(
            o2, (long)H1, 0,
            blk_w + (long)i * H1 * H1, (long)H1, 0,
            blk_b + (long)i * H1, 0,
            pre, (long)H1, 0, H1);
        ln_lrelu_kernel<<<dim3((unsigned)M), blk, 0, stream>>>(
            pre, o2, blk_ln_g + (long)i * H1, blk_ln_b + (long)i * H1, H1);
    }

    // fc3: [M x H1] @ [H1 x H2]
    gemm_wmma_f32<32><<<dim3(H2 / 64, M / 128, 1), blk, 0, stream>>>(
        o2, (long)H1, 0, fc3_w, (long)H1, 0, fc3_b, 0, o3, (long)H2, 0, H1);
    ln_lrelu_kernel<<<dim3((unsigned)M), blk, 0, stream>>>(o3, o3, ln3_g, ln3_b, H2);

    // fc4: [M x H2] @ [H2 x 3]
    fc4_kernel<<<dim3((unsigned)M), blk, 0, stream>>>(o3, fc4_w, fc4_b, o4, H2);
}